// VolumetricCELoss_75720273428985
// MI455X (gfx1250) — compile-verified
//
#include <hip/hip_runtime.h>
#include <hip/hip_bf16.h>
#include <math.h>

// ---------------------------------------------------------------------------
// VolumetricCELoss for MI455X (gfx1250, wave32)
//
// Phase 1: nearest-voxel argmin. Joints 0..15 via V_WMMA_F32_16X16X4_F32:
//   A (16 voxels x K=4) = (x, y, z, |c|^2) ; B (4 x 16 joints) = (-2k, 1)
//   => D[v,j] = |c|^2 - 2 c.k  (argmin over v identical to full distance).
//   Joint 16 handled with 3 per-lane FMAs (each lane already holds a voxel),
//   replacing a 1/16-utilized second WMMA tile + its 24-VALU epilogue.
//   Epilogue keeps 8 independent per-row running minima (no serial chain),
//   tracking the loop-uniform tile base instead of per-element indices.
// Phase 2: deterministic 2-stage sum of volumes_batch_pred (float4, fixed
//   34-trip loop: NPRED/4 == 34 * 1024 * 256 exactly).
// Phase 3: single block: total, 3x3x3 clamped window sums, log-loss, scalar.
// ---------------------------------------------------------------------------

typedef float v2f __attribute__((ext_vector_type(2)));
typedef float v8f __attribute__((ext_vector_type(8)));

#define B_  32
#define J_  17
#define D_  16
#define H_  64
#define W_  64
#define V_  (D_ * H_ * W_)           /* 65536 */
#define NPRED ((size_t)B_ * J_ * V_) /* 35651584 */

#define P1_BLOCKS_PER_BATCH 16
#define P1_WAVES_PER_BLOCK  8
#define P1_VOX_PER_WAVE     (V_ / (P1_BLOCKS_PER_BATCH * P1_WAVES_PER_BLOCK)) /* 512 */

#define SUM_BLOCKS  1024
#define SUM_THREADS 256
#define SUM_ITERS   34   /* NPRED/4 / (SUM_BLOCKS*SUM_THREADS) == 34 exactly */

// ---------------------------------------------------------------- init ------
__global__ void init_argmin_kernel(unsigned long long* __restrict__ argmin) {
    int i = blockIdx.x * blockDim.x + threadIdx.x;
    if (i < B_ * J_) argmin[i] = 0xFFFFFFFFFFFFFFFFull;
}

__device__ __forceinline__ unsigned long long pack_key(float v, unsigned idx) {
    unsigned fb  = __float_as_uint(v);
    unsigned key = fb ^ ((fb >> 31) ? 0xFFFFFFFFu : 0x80000000u);
    return ((unsigned long long)key << 32) | (unsigned long long)idx;
}

// ------------------------------------------------------ phase 1: argmin -----
__global__ __launch_bounds__(256) void argmin_wmma_kernel(
    const float* __restrict__ coords,            // [B, V, 3]
    const float* __restrict__ kps,               // [B, J, 3]
    unsigned long long* __restrict__ argmin) {   // [B*J] packed (key<<32)|idx
    const int lane = threadIdx.x & 31;
    const int wave = threadIdx.x >> 5;
    const int b     = blockIdx.x / P1_BLOCKS_PER_BATCH;
    const int chunk = blockIdx.x % P1_BLOCKS_PER_BATCH;
    const int col   = lane & 15;        // joint column / voxel slot within tile
    const bool hi   = lane >= 16;       // upper half-wave

    // B matrix for joints 0..15.
    // Layout (32-bit B 4x16): lane<16 holds K=0,1; lane>=16 holds K=2,3.
    v2f bm;
    {
        const float* kp = kps + ((size_t)b * J_ + col) * 3;
        const float kx = kp[0], ky = kp[1], kz = kp[2];
        // column j of B = (-2kx, -2ky, -2kz, 1)
        bm[0] = hi ? (-2.f * kz) : (-2.f * kx);
        bm[1] = hi ? 1.0f        : (-2.f * ky);
    }
    // Joint 16: broadcast keypoint, scaled by -2 (uniform across lanes).
    float n16x, n16y, n16z;
    {
        const float* kp = kps + ((size_t)b * J_ + 16) * 3;
        n16x = -2.f * kp[0]; n16y = -2.f * kp[1]; n16z = -2.f * kp[2];
    }

    // 8 independent running minima for the WMMA tile + 1 for joint 16.
    float rminv[8];
    int   rbase[8];
#pragma unroll
    for (int r = 0; r < 8; ++r) { rminv[r] = 3.402823466e+38f; rbase[r] = 0; }
    float min16  = 3.402823466e+38f;
    int   base16 = 0;

    const int   waveBase = chunk * (P1_WAVES_PER_BLOCK * P1_VOX_PER_WAVE) +
                           wave * P1_VOX_PER_WAVE;
    const float* cb = coords + (size_t)b * V_ * 3;

    for (int it = 0; it < P1_VOX_PER_WAVE / 16; ++it) {
        const int tileBase = waveBase + it * 16;     // loop-uniform
        const int v = tileBase + col;                // both halves read same 16 voxels
        const float* c = cb + (size_t)v * 3;
        const float x = c[0], y = c[1], z = c[2];
        const float c2 = fmaf(x, x, fmaf(y, y, z * z));
        // A layout (32-bit A 16x4): lane<16 -> K=0,1 ; lane>=16 -> K=2,3.
        v2f a;
        a[0] = hi ? z  : x;
        a[1] = hi ? c2 : y;

        v8f cz = {0.f, 0.f, 0.f, 0.f, 0.f, 0.f, 0.f, 0.f};
        v8f d = __builtin_amdgcn_wmma_f32_16x16x4_f32(
            false, a, false, bm, (short)0, cz, false, false);

        // Joint 16: per-lane scalar distance surrogate for this lane's voxel.
        const float d16 = fmaf(x, n16x, fmaf(y, n16y, fmaf(z, n16z, c2)));
        if (d16 < min16) { min16 = d16; base16 = tileBase; }

#pragma unroll
        for (int r = 0; r < 8; ++r) {
            const float dv = d[r];
            if (dv < rminv[r]) { rminv[r] = dv; rbase[r] = tileBase; }
        }
    }

    // Merge 8 rows with tie-correct packed u64 keys, then atomic per joint.
    const int rowAdd = hi ? 8 : 0;
    {
        unsigned long long best = 0xFFFFFFFFFFFFFFFFull;
#pragma unroll
        for (int r = 0; r < 8; ++r) {
            unsigned long long p = pack_key(rminv[r], (unsigned)(rbase[r] + r + rowAdd));
            best = p < best ? p : best;
        }
        atomicMin(&argmin[b * J_ + col], best);   // joints 0..15 (all lanes valid)
    }
    // Joint 16: every lane holds an independent partial min (halves duplicate).
    atomicMin(&argmin[b * J_ + 16], pack_key(min16, (unsigned)(base16 + col)));
}

// ------------------------------------------------ phase 2: partial sums ----
__global__ __launch_bounds__(SUM_THREADS) void sum_partial_kernel(
    const float4* __restrict__ pred4, float* __restrict__ partials) {
    __shared__ float sm[SUM_THREADS];
    const size_t base   = (size_t)blockIdx.x * SUM_THREADS + threadIdx.x;
    const size_t stride = (size_t)SUM_BLOCKS * SUM_THREADS;
    float s0 = 0.f, s1 = 0.f;
#pragma unroll 2
    for (int k = 0; k < SUM_ITERS; k += 2) {
        float4 a = pred4[base + (size_t)k * stride];
        float4 b = pred4[base + (size_t)(k + 1) * stride];
        s0 += (a.x + a.y) + (a.z + a.w);
        s1 += (b.x + b.y) + (b.z + b.w);
    }
    sm[threadIdx.x] = s0 + s1;
    __syncthreads();
    for (int off = SUM_THREADS / 2; off > 0; off >>= 1) {
        if ((int)threadIdx.x < off) sm[threadIdx.x] += sm[threadIdx.x + off];
        __syncthreads();
    }
    if (threadIdx.x == 0) partials[blockIdx.x] = sm[0];
}

// ----------------------------------------------------- phase 3: finalize ---
__global__ __launch_bounds__(256) void finalize_kernel(
    const float* __restrict__ pred,                 // [B,J,D,H,W]
    const float* __restrict__ validity,             // [B,J,1]
    const unsigned long long* __restrict__ argmin,  // [B*J]
    const float* __restrict__ partials,             // [SUM_BLOCKS]
    float* __restrict__ out) {
    __shared__ float sm[256];
    __shared__ float totalSh;

    float s = 0.f;
    for (int i = threadIdx.x; i < SUM_BLOCKS; i += 256) s += partials[i];
    sm[threadIdx.x] = s;
    __syncthreads();
    for (int off = 128; off > 0; off >>= 1) {
        if ((int)threadIdx.x < off) sm[threadIdx.x] += sm[threadIdx.x + off];
        __syncthreads();
    }
    if (threadIdx.x == 0) totalSh = sm[0];
    __syncthreads();
    const float total = totalSh;

    float acc = 0.f;
    for (int p = threadIdx.x; p < B_ * J_; p += 256) {
        const unsigned idx = (unsigned)(argmin[p] & 0xFFFFFFFFull);
        const int iz = (int)(idx >> 12);
        const int iy = (int)((idx >> 6) & 63);
        const int ix = (int)(idx & 63);
        const float* vp = pred + (size_t)p * V_;
        float focus = 0.f;
        for (int dz = -1; dz <= 1; ++dz) {
            const int z = iz + dz;
            if (z < 0 || z >= D_) continue;
            for (int dy = -1; dy <= 1; ++dy) {
                const int y = iy + dy;
                if (y < 0 || y >= H_) continue;
                for (int dx = -1; dx <= 1; ++dx) {
                    const int x = ix + dx;
                    if (x < 0 || x >= W_) continue;
                    focus += vp[(z * H_ + y) * W_ + x];
                }
            }
        }
        const float val = validity[p];
        acc += val * (-logf(focus + 1e-6f) + logf(total - focus));
    }
    __syncthreads();
    sm[threadIdx.x] = acc;
    __syncthreads();
    for (int off = 128; off > 0; off >>= 1) {
        if ((int)threadIdx.x < off) sm[threadIdx.x] += sm[threadIdx.x + off];
        __syncthreads();
    }
    if (threadIdx.x == 0) out[0] = sm[0] / (float)(B_ * J_);
}

// --------------------------------------------------------------- launch ----
extern "C" void kernel_launch(void* const* d_in, const int* in_sizes, int n_in,
                              void* d_out, int out_size, void* d_ws, size_t ws_size,
                              hipStream_t stream) {
    const float* coords   = (const float*)d_in[0];  // [B,D,H,W,3]
    const float* pred     = (const float*)d_in[1];  // [B,J,D,H,W]
    const float* kps      = (const float*)d_in[2];  // [B,J,3]
    const float* validity = (const float*)d_in[3];  // [B,J,1]
    float* out = (float*)d_out;

    // Workspace layout: argmin[B*J] u64 (4352 B), then partials[SUM_BLOCKS] f32.
    unsigned long long* ws_argmin = (unsigned long long*)d_ws;
    float* ws_partials = (float*)((char*)d_ws + (size_t)B_ * J_ * sizeof(unsigned long long));

    init_argmin_kernel<<<(B_ * J_ + 255) / 256, 256, 0, stream>>>(ws_argmin);

    argmin_wmma_kernel<<<B_ * P1_BLOCKS_PER_BATCH, 256, 0, stream>>>(
        coords, kps, ws_argmin);

    sum_partial_kernel<<<SUM_BLOCKS, SUM_THREADS, 0, stream>>>(
        (const float4*)pred, ws_partials);

    finalize_kernel<<<1, 256, 0, stream>>>(pred, validity, ws_argmin,
                                           ws_partials, out);
}